// LinearPredictiveCoefficientsStabilityCheck_72842645340721
// MI455X (gfx1250) — compile-verified
//
#include <hip/hip_runtime.h>
#include <stdint.h>

// LPC stability check: a -> (lpc2par) -> clip(+-1) -> (par2lpc) -> out
// Rows of W=25 fp32; one row per lane, staged through LDS with CDNA5
// async global<->LDS DMA (ASYNCcnt path).

#define ORDER 24
#define W 25
#define ROWS 256                         // rows (threads) per block
#define TILE_ELEMS (ROWS * W)            // 6400 floats
#define TILE_B128 ((TILE_ELEMS * 4) / 16) // 1600 x 16B transfers

typedef __attribute__((address_space(3))) float lds_f32;

__global__ __launch_bounds__(ROWS)
void lpc_stab_kernel(const float* __restrict__ a,
                     float* __restrict__ out,
                     int nRows)
{
    __shared__ float tile[TILE_ELEMS];   // 25600 B

    const int tid = threadIdx.x;
    const long long row0 = (long long)blockIdx.x * ROWS;
    const long long rem  = (long long)nRows - row0;
    const bool full = (rem >= (long long)ROWS);

    const float* gin  = a   + row0 * W;
    float*       gout = out + row0 * W;

    // Raw LDS byte offset of the tile (AS3 pointers are 32-bit offsets).
    const unsigned ldsBase = (unsigned)(unsigned long long)(lds_f32*)tile;

    // ---------------- stage in: global -> LDS (async DMA) ----------------
    if (full) {
#pragma unroll
        for (int it = 0; it < 7; ++it) {
            int idx = it * ROWS + tid;            // b128 transfer index
            if (idx < TILE_B128) {
                unsigned voff  = (unsigned)idx * 16u; // byte offset in tile
                unsigned laddr = ldsBase + voff;
                asm volatile("global_load_async_to_lds_b128 %0, %1, %2"
                             :: "v"(laddr), "v"(voff), "s"(gin)
                             : "memory");
            }
        }
        asm volatile("s_wait_asynccnt 0x0" ::: "memory");
    } else {
        int n = (int)rem * W;
        for (int j = tid; j < n; j += ROWS) tile[j] = gin[j];
    }
    __syncthreads();

    // ---------------- per-row recursion, fully in registers ----------------
    const bool active = (row0 + (long long)tid) < (long long)nRows;
    if (active) {
        float c[W];
#pragma unroll
        for (int j = 0; j < W; ++j) c[j] = tile[tid * W + j];
        // c[0] = K passes through untouched; c[1..24] = LPC coeffs.

        // ---- lpc2par: downward Levinson recursion, in place ----
        // Step m leaves c[1+m] untouched, so the array ends as [K, k0..k23].
#pragma unroll
        for (int m = ORDER - 1; m >= 1; --m) {
            float km = c[1 + m];
            float r  = 1.0f / (1.0f - km * km);
#pragma unroll
            for (int i = 0; i < (m + 1) / 2; ++i) {
                int jp = m - 1 - i;               // mirror index
                float xi = c[1 + i];
                float xj = c[1 + jp];
                c[1 + i] = (xi - km * xj) * r;
                if (i != jp) c[1 + jp] = (xj - km * xi) * r;
            }
        }

        // ---- clip: fp32(1.0 - 1e-16) == 1.0f exactly ----
#pragma unroll
        for (int j = 1; j < W; ++j)
            c[j] = fminf(fmaxf(c[j], -1.0f), 1.0f);

        // ---- par2lpc: upward recursion, in place ----
        // Before step m, c[1..m] holds a[0..m-1]; c[1+m] already holds km.
#pragma unroll
        for (int m = 1; m <= ORDER - 1; ++m) {
            float km = c[1 + m];
#pragma unroll
            for (int i = 0; i < (m + 1) / 2; ++i) {
                int jp = m - 1 - i;
                float xi = c[1 + i];
                float xj = c[1 + jp];
                c[1 + i] = xi + km * xj;
                if (i != jp) c[1 + jp] = xj + km * xi;
            }
        }

#pragma unroll
        for (int j = 0; j < W; ++j) tile[tid * W + j] = c[j];
    }
    __syncthreads();

    // ---------------- stage out: LDS -> global (async DMA) ----------------
    if (full) {
#pragma unroll
        for (int it = 0; it < 7; ++it) {
            int idx = it * ROWS + tid;
            if (idx < TILE_B128) {
                unsigned voff  = (unsigned)idx * 16u;
                unsigned laddr = ldsBase + voff;
                asm volatile("global_store_async_from_lds_b128 %0, %1, %2"
                             :: "v"(voff), "v"(laddr), "s"(gout)
                             : "memory");
            }
        }
        asm volatile("s_wait_asynccnt 0x0" ::: "memory");
    } else {
        int n = (int)rem * W;
        for (int j = tid; j < n; j += ROWS) gout[j] = tile[j];
    }
}

extern "C" void kernel_launch(void* const* d_in, const int* in_sizes, int n_in,
                              void* d_out, int out_size, void* d_ws, size_t ws_size,
                              hipStream_t stream) {
    (void)n_in; (void)d_ws; (void)ws_size; (void)out_size;
    const float* a = (const float*)d_in[0];
    float* out = (float*)d_out;
    const int nRows = in_sizes[0] / W;          // 512*4096 = 2,097,152
    const int blocks = (nRows + ROWS - 1) / ROWS;
    lpc_stab_kernel<<<blocks, ROWS, 0, stream>>>(a, out, nRows);
}